// SchNetLayer_72086731096588
// MI455X (gfx1250) — compile-verified
//
#include <hip/hip_runtime.h>

#define N_NODES 10000
#define E_EDGES 320000
#define HDIM    256
#define NG      20
#define BN_EPS  1e-5f

typedef float v2f __attribute__((ext_vector_type(2)));
typedef float v8f __attribute__((ext_vector_type(8)));
typedef unsigned int u32x4_t __attribute__((ext_vector_type(4)));
typedef int i32x8_t __attribute__((ext_vector_type(8)));
typedef int i32x4_t __attribute__((ext_vector_type(4)));

// Fast SiLU: z * v_rcp_f32(1 + e^-z)  (avoids the IEEE div_scale/NR chain)
__device__ __forceinline__ float silu_f(float z) {
    return z * __builtin_amdgcn_rcpf(1.0f + __expf(-z));
}

__device__ __forceinline__ v8f wmma4(v2f a, v2f b, v8f c) {
    return __builtin_amdgcn_wmma_f32_16x16x4_f32(false, a, false, b, (short)0, c,
                                                 false, false);
}

// ---------------------------------------------------------------------------
// Tensor Data Mover: DMA a 2D f32 tile (tile_d0 elems/row x tile_d1 rows) from
// global into LDS. D# layout per CDNA5 ISA 8.3-8.5: group0 {count=1, lds_addr,
// global_addr, type=2}, group1 {data_size=4B, tensor dims, tile dims, stride}.
// OOB rows (beyond tensor_d1) read as zero. Issue from ONE wave per block.
// This toolchain exposes the 6-arg builtin:
//   (u32x4 g0, i32x8 g1, i32x4 g2, i32x4 g3, i32x8 extra, i32 cpol)
// ---------------------------------------------------------------------------
#if __has_builtin(__builtin_amdgcn_tensor_load_to_lds) && \
    __has_builtin(__builtin_amdgcn_s_wait_tensorcnt)
#define USE_TDM 1
#else
#define USE_TDM 0
#endif

#if USE_TDM
__device__ __forceinline__ void tdm_load_tile_2d_f32(
    void* lds_dst, const float* global_tile,
    unsigned tile_d0, unsigned tile_d1,
    unsigned tensor_d0, unsigned tensor_d1,
    unsigned row_stride_elems)
{
    union { unsigned long long q[2]; u32x4_t v; } g0;
    union { unsigned long long q[4]; i32x8_t v; } g1;
    const unsigned long long gaddr = (unsigned long long)(uintptr_t)global_tile;
    const unsigned long long lds   = (unsigned)(uintptr_t)lds_dst;

    g0.q[0] = 1ull | (lds << 32);                          // count=1 | lds_addr
    g0.q[1] = (gaddr & ((1ull << 57) - 1)) | (2ull << 62); // global_addr | type=2

    const unsigned long long d0 = tensor_d0, d1 = tensor_d1;
    const unsigned long long st = row_stride_elems;
    g1.q[0] = (2ull << 16)                                  // data_size = 2 -> 4B
            | ((d0 & 0xFFFFull) << 48);                     // tensor_dim0[15:0]
    g1.q[1] = (d0 >> 16)                                    // tensor_dim0[31:16]
            | (d1 << 16)                                    // tensor_dim1
            | ((unsigned long long)(tile_d0 & 0xFFFF) << 48); // tile_dim0
    g1.q[2] = (unsigned long long)(tile_d1 & 0xFFFF)        // tile_dim1
            | ((st & 0xFFFFFFFFull) << 32);                 // dim0_stride[31:0]
    g1.q[3] = (st >> 32) & 0xFFFFull;                       // dim0_stride[47:32]

    const i32x4_t z4 = {0, 0, 0, 0};
    const i32x8_t z8 = {0, 0, 0, 0, 0, 0, 0, 0};
    __builtin_amdgcn_tensor_load_to_lds(g0.v, g1.v, z4, z4, z8, 0);
}
#endif

// ---------------------------------------------------------------------------
// GEMM epilogue helpers (C/D layout: VGPR v -> row v + 8*(lane>=16))
// ---------------------------------------------------------------------------
__device__ __forceinline__ void epilog_silu_lds(float (*dst)[HDIM], int mrow,
                                                int col, v8f acc,
                                                const float* __restrict__ bias) {
    const float bb = bias[col];
#pragma unroll
    for (int v = 0; v < 8; ++v) dst[mrow + v][col] = silu_f(acc[v] + bb);
}

__device__ __forceinline__ void epilog_bias_lds(float (*dst)[HDIM], int mrow,
                                                int col, v8f acc,
                                                const float* __restrict__ bias) {
    const float bb = bias[col];
#pragma unroll
    for (int v = 0; v < 8; ++v) dst[mrow + v][col] = acc[v] + bb;
}

__device__ __forceinline__ void epilog_bias_global(float* __restrict__ dst,
                                                   int n0, int mrow, int col,
                                                   v8f acc,
                                                   const float* __restrict__ bias) {
    const float bb = bias[col];
#pragma unroll
    for (int v = 0; v < 8; ++v) {
        const int node = n0 + mrow + v;
        if (node < N_NODES) dst[(size_t)node * HDIM + col] = acc[v] + bb;
    }
}

// ---------------------------------------------------------------------------
// Kernel 0: zero aggregation buffer (atomics accumulate into it every call)
// ---------------------------------------------------------------------------
__global__ void zero_kernel(float* __restrict__ p, size_t n) {
    const size_t stride = (size_t)gridDim.x * blockDim.x;
    for (size_t i = (size_t)blockIdx.x * blockDim.x + threadIdx.x; i < n; i += stride)
        p[i] = 0.0f;
}

// ---------------------------------------------------------------------------
// Kernel 1: fused edge pipeline, 32 edges/block, 8 waves.
// Wave w owns M-tile (w>>2) x 4 adjacent N-tiles at column (w&3)*64:
// A fragment loaded once per K-step, reused by 4 WMMAs (4 live accumulators).
// ---------------------------------------------------------------------------
__global__ __launch_bounds__(256) void edge_kernel(
    const float* __restrict__ x, const float* __restrict__ rbf,
    const int* __restrict__ ei,
    const float* __restrict__ Wf1, const float* __restrict__ bf1,
    const float* __restrict__ Wf2, const float* __restrict__ bf2,
    float* __restrict__ agg)
{
    __shared__ float s_rbf[32][NG];
    __shared__ float s_t1[32][HDIM];
    __shared__ float s_filt[32][HDIM];
    __shared__ int s_row[32], s_col[32];

    const int tid  = threadIdx.x;
    const int lane = tid & 31;
    const int wave = tid >> 5;
    const int half = lane >> 4;
    const int l16  = lane & 15;
    const int e0   = blockIdx.x * 32;

    if (tid < 32) {
        s_row[tid] = ei[e0 + tid];            // ei[0][e]
        s_col[tid] = ei[E_EDGES + e0 + tid];  // ei[1][e]
    }
#if USE_TDM
    if (wave == 0) {
        tdm_load_tile_2d_f32(&s_rbf[0][0], rbf + (size_t)e0 * NG,
                             NG, 32, NG, 32, NG);
        __builtin_amdgcn_s_wait_tensorcnt(0);
    }
#else
    for (int i = tid; i < 32 * NG; i += 256)
        s_rbf[i / NG][i % NG] = rbf[(size_t)e0 * NG + i];
#endif
    __syncthreads();

    const int mt   = wave >> 2;             // 0..1
    const int nb   = (wave & 3) * 64 + l16; // base column for this lane
    const int am   = mt * 16 + l16;         // A row for this lane
    const int mrow = mt * 16 + half * 8;    // D base row for this lane half

    // ---- stage 1: t1 = SiLU(rbf @ Wf1 + bf1), K = 20 ----
    {
        v8f acc0 = {}, acc1 = {}, acc2 = {}, acc3 = {};
#pragma unroll
        for (int k0 = 0; k0 < NG; k0 += 4) {
            const int ak = k0 + half * 2;
            v2f a;
            a[0] = s_rbf[am][ak];
            a[1] = s_rbf[am][ak + 1];
            const float* B0 = Wf1 + (size_t)ak * HDIM + nb;
            const float* B1 = B0 + HDIM;
            v2f b;
            b[0] = B0[0];  b[1] = B1[0];  acc0 = wmma4(a, b, acc0);
            b[0] = B0[16]; b[1] = B1[16]; acc1 = wmma4(a, b, acc1);
            b[0] = B0[32]; b[1] = B1[32]; acc2 = wmma4(a, b, acc2);
            b[0] = B0[48]; b[1] = B1[48]; acc3 = wmma4(a, b, acc3);
        }
        epilog_silu_lds(s_t1, mrow, nb,      acc0, bf1);
        epilog_silu_lds(s_t1, mrow, nb + 16, acc1, bf1);
        epilog_silu_lds(s_t1, mrow, nb + 32, acc2, bf1);
        epilog_silu_lds(s_t1, mrow, nb + 48, acc3, bf1);
    }
    __syncthreads();

    // ---- stage 2: filt = t1 @ Wf2 + bf2, K = 256 ----
    {
        v8f acc0 = {}, acc1 = {}, acc2 = {}, acc3 = {};
        for (int k0 = 0; k0 < HDIM; k0 += 4) {
            const int ak = k0 + half * 2;
            v2f a;
            a[0] = s_t1[am][ak];
            a[1] = s_t1[am][ak + 1];
            const float* B0 = Wf2 + (size_t)ak * HDIM + nb;
            const float* B1 = B0 + HDIM;
            v2f b;
            b[0] = B0[0];  b[1] = B1[0];  acc0 = wmma4(a, b, acc0);
            b[0] = B0[16]; b[1] = B1[16]; acc1 = wmma4(a, b, acc1);
            b[0] = B0[32]; b[1] = B1[32]; acc2 = wmma4(a, b, acc2);
            b[0] = B0[48]; b[1] = B1[48]; acc3 = wmma4(a, b, acc3);
        }
        epilog_bias_lds(s_filt, mrow, nb,      acc0, bf2);
        epilog_bias_lds(s_filt, mrow, nb + 16, acc1, bf2);
        epilog_bias_lds(s_filt, mrow, nb + 32, acc2, bf2);
        epilog_bias_lds(s_filt, mrow, nb + 48, acc3, bf2);
    }
    __syncthreads();

    // ---- gather x[col] * filt, scatter-add into agg[row] ----
    for (int i = tid; i < 32 * HDIM; i += 256) {
        const int el = i >> 8;
        const int f  = i & (HDIM - 1);
        const int r = s_row[el];
        const int c = s_col[el];
        const float m = x[(size_t)c * HDIM + f] * s_filt[el][f];
        unsafeAtomicAdd(&agg[(size_t)r * HDIM + f], m);
    }
}

// ---------------------------------------------------------------------------
// Kernel 2: node update  h = SiLU(agg@Wu1+bu1)@Wu2+bu2, 32 nodes/block.
// agg tile pulled in by the Tensor Data Mover; OOB tail rows read as zero.
// ---------------------------------------------------------------------------
__global__ __launch_bounds__(256) void node_kernel(
    const float* __restrict__ agg,
    const float* __restrict__ Wu1, const float* __restrict__ bu1,
    const float* __restrict__ Wu2, const float* __restrict__ bu2,
    float* __restrict__ h)
{
    __shared__ float s_a[32][HDIM];
    __shared__ float s_t1[32][HDIM];

    const int tid  = threadIdx.x;
    const int lane = tid & 31;
    const int wave = tid >> 5;
    const int half = lane >> 4;
    const int l16  = lane & 15;
    const int n0   = blockIdx.x * 32;

#if USE_TDM
    if (wave == 0) {
        const unsigned rows_left = (unsigned)(N_NODES - n0);  // OOB rows -> 0
        tdm_load_tile_2d_f32(&s_a[0][0], agg + (size_t)n0 * HDIM,
                             HDIM, 32, HDIM, rows_left, HDIM);
        __builtin_amdgcn_s_wait_tensorcnt(0);
    }
#else
    for (int i = tid; i < 32 * HDIM; i += 256) {
        const int r = i >> 8, f = i & (HDIM - 1);
        const int node = n0 + r;
        s_a[r][f] = (node < N_NODES) ? agg[(size_t)node * HDIM + f] : 0.0f;
    }
#endif
    __syncthreads();

    const int mt   = wave >> 2;
    const int nb   = (wave & 3) * 64 + l16;
    const int am   = mt * 16 + l16;
    const int mrow = mt * 16 + half * 8;

    // ---- stage 1: t1 = SiLU(agg @ Wu1 + bu1), K = 256 ----
    {
        v8f acc0 = {}, acc1 = {}, acc2 = {}, acc3 = {};
        for (int k0 = 0; k0 < HDIM; k0 += 4) {
            const int ak = k0 + half * 2;
            v2f a;
            a[0] = s_a[am][ak];
            a[1] = s_a[am][ak + 1];
            const float* B0 = Wu1 + (size_t)ak * HDIM + nb;
            const float* B1 = B0 + HDIM;
            v2f b;
            b[0] = B0[0];  b[1] = B1[0];  acc0 = wmma4(a, b, acc0);
            b[0] = B0[16]; b[1] = B1[16]; acc1 = wmma4(a, b, acc1);
            b[0] = B0[32]; b[1] = B1[32]; acc2 = wmma4(a, b, acc2);
            b[0] = B0[48]; b[1] = B1[48]; acc3 = wmma4(a, b, acc3);
        }
        epilog_silu_lds(s_t1, mrow, nb,      acc0, bu1);
        epilog_silu_lds(s_t1, mrow, nb + 16, acc1, bu1);
        epilog_silu_lds(s_t1, mrow, nb + 32, acc2, bu1);
        epilog_silu_lds(s_t1, mrow, nb + 48, acc3, bu1);
    }
    __syncthreads();

    // ---- stage 2: h = t1 @ Wu2 + bu2, K = 256, store to global ----
    {
        v8f acc0 = {}, acc1 = {}, acc2 = {}, acc3 = {};
        for (int k0 = 0; k0 < HDIM; k0 += 4) {
            const int ak = k0 + half * 2;
            v2f a;
            a[0] = s_t1[am][ak];
            a[1] = s_t1[am][ak + 1];
            const float* B0 = Wu2 + (size_t)ak * HDIM + nb;
            const float* B1 = B0 + HDIM;
            v2f b;
            b[0] = B0[0];  b[1] = B1[0];  acc0 = wmma4(a, b, acc0);
            b[0] = B0[16]; b[1] = B1[16]; acc1 = wmma4(a, b, acc1);
            b[0] = B0[32]; b[1] = B1[32]; acc2 = wmma4(a, b, acc2);
            b[0] = B0[48]; b[1] = B1[48]; acc3 = wmma4(a, b, acc3);
        }
        epilog_bias_global(h, n0, mrow, nb,      acc0, bu2);
        epilog_bias_global(h, n0, mrow, nb + 16, acc1, bu2);
        epilog_bias_global(h, n0, mrow, nb + 32, acc2, bu2);
        epilog_bias_global(h, n0, mrow, nb + 48, acc3, bu2);
    }
}

// ---------------------------------------------------------------------------
// Kernel 3: BatchNorm statistics. One block per feature, LDS tree reduction.
// ---------------------------------------------------------------------------
__global__ __launch_bounds__(256) void bn_stats_kernel(
    const float* __restrict__ h, float* __restrict__ mean,
    float* __restrict__ invstd)
{
    __shared__ float rs[256], rss[256];
    const int f = blockIdx.x;
    float s = 0.0f, ss = 0.0f;
    for (int n = threadIdx.x; n < N_NODES; n += 256) {
        const float v = h[(size_t)n * HDIM + f];
        s += v; ss += v * v;
    }
    rs[threadIdx.x] = s; rss[threadIdx.x] = ss;
    __syncthreads();
    for (int off = 128; off > 0; off >>= 1) {
        if (threadIdx.x < off) {
            rs[threadIdx.x]  += rs[threadIdx.x + off];
            rss[threadIdx.x] += rss[threadIdx.x + off];
        }
        __syncthreads();
    }
    if (threadIdx.x == 0) {
        const float mu  = rs[0] / (float)N_NODES;
        const float var = rss[0] / (float)N_NODES - mu * mu;
        mean[f]   = mu;
        invstd[f] = rsqrtf(var + BN_EPS);
    }
}

// ---------------------------------------------------------------------------
// Kernel 4: out = x + gamma*(h-mean)*invstd + beta
// ---------------------------------------------------------------------------
__global__ void bn_apply_kernel(
    const float* __restrict__ x, const float* __restrict__ h,
    const float* __restrict__ gamma, const float* __restrict__ beta,
    const float* __restrict__ mean, const float* __restrict__ invstd,
    float* __restrict__ out)
{
    const size_t total = (size_t)N_NODES * HDIM;
    const size_t stride = (size_t)gridDim.x * blockDim.x;
    for (size_t i = (size_t)blockIdx.x * blockDim.x + threadIdx.x; i < total; i += stride) {
        const int f = (int)(i & (HDIM - 1));
        out[i] = x[i] + gamma[f] * (h[i] - mean[f]) * invstd[f] + beta[f];
    }
}

// ---------------------------------------------------------------------------
extern "C" void kernel_launch(void* const* d_in, const int* in_sizes, int n_in,
                              void* d_out, int out_size, void* d_ws, size_t ws_size,
                              hipStream_t stream) {
    const float* x     = (const float*)d_in[0];
    const float* rbf   = (const float*)d_in[1];
    const int*   ei    = (const int*)d_in[2];
    const float* Wf1   = (const float*)d_in[3];
    const float* bf1   = (const float*)d_in[4];
    const float* Wf2   = (const float*)d_in[5];
    const float* bf2   = (const float*)d_in[6];
    const float* Wu1   = (const float*)d_in[7];
    const float* bu1   = (const float*)d_in[8];
    const float* Wu2   = (const float*)d_in[9];
    const float* bu2   = (const float*)d_in[10];
    const float* gamma = (const float*)d_in[11];
    const float* beta  = (const float*)d_in[12];
    float* out = (float*)d_out;

    float* agg    = (float*)d_ws;                       // N*HDIM
    float* h      = agg + (size_t)N_NODES * HDIM;       // N*HDIM
    float* mean   = h   + (size_t)N_NODES * HDIM;       // HDIM
    float* invstd = mean + HDIM;                        // HDIM

    zero_kernel<<<2048, 256, 0, stream>>>(agg, (size_t)N_NODES * HDIM);
    edge_kernel<<<E_EDGES / 32, 256, 0, stream>>>(x, rbf, ei, Wf1, bf1, Wf2, bf2, agg);
    node_kernel<<<(N_NODES + 31) / 32, 256, 0, stream>>>(agg, Wu1, bu1, Wu2, bu2, h);
    bn_stats_kernel<<<HDIM, 256, 0, stream>>>(h, mean, invstd);
    bn_apply_kernel<<<2048, 256, 0, stream>>>(x, h, gamma, beta, mean, invstd, out);
}